// PartialFC_40484361732593
// MI455X (gfx1250) — compile-verified
//
#include <hip/hip_runtime.h>
#include <hip/hip_bf16.h>

// ---------------------------------------------------------------------------
// PartialFC sampled-softmax loss for MI455X (gfx1250, wave32, WMMA).
//   inputs : features[512,512] f32, weight[500000,512] f32,
//            labels[512] i32, index[50000] i32 (sorted, contains all labels)
//   output : logits[512,50000] f32  ++  loss (1 f32)
// Strategy: inv-norms only in d_ws (200KB); GEMM gathers weight rows directly
// (102MB working set is L2-resident on the 192MB L2). The per-column
// normalization scale is hoisted OUT of the K-loop and applied once to the
// accumulator (D-layout: all 8 acc VGPRs of a lane share one N column).
// fp32 matrix math via V_WMMA_F32_16X16X4_F32 to match reference precision.
// ---------------------------------------------------------------------------

#define BATCH   512
#define EMBED   512
#define NSAMP   50000

typedef __attribute__((ext_vector_type(2))) float v2f;
typedef __attribute__((ext_vector_type(8))) float v8f;

// ---------------- Kernel 1: inverse row norms of gathered weight rows ------
__global__ __launch_bounds__(256) void k_norms(const float* __restrict__ weight,
                                               const int* __restrict__ index,
                                               float* __restrict__ inv_norm) {
    int s = blockIdx.x * 8 + (threadIdx.x >> 5);   // one wave32 per sampled row
    if (s >= NSAMP) return;
    int lane = threadIdx.x & 31;
    const float* row = weight + (size_t)index[s] * EMBED;
    float ss = 0.f;
    // 32 lanes x float4 -> 512B fully-coalesced per step, 4 steps per row
    for (int k = lane * 4; k < EMBED; k += 32 * 4) {
        float4 v = *(const float4*)(row + k);
        ss += v.x * v.x + v.y * v.y + v.z * v.z + v.w * v.w;
    }
    for (int off = 16; off > 0; off >>= 1) ss += __shfl_xor(ss, off, 32);
    if (lane == 0) inv_norm[s] = 1.0f / fmaxf(sqrtf(ss), 1e-12f);
}

// ---------------- Kernel 2: logits = F @ norm_W^T via fp32 WMMA ------------
// Block: 256 thr = 8 waves. Block tile: M=16 (LDS-staged features), N=128.
// Wave tile: 16x16, K-loop 512 in steps of 8 with two accumulator chains.
// Inner loop is pure load->wmma; normalization applied once at epilogue.
__global__ __launch_bounds__(256) void k_gemm(const float* __restrict__ features,
                                              const float* __restrict__ weight,
                                              const int* __restrict__ index,
                                              const float* __restrict__ inv_norm,
                                              float* __restrict__ logits) {
    __shared__ float Af[16 * EMBED];               // 32 KB of 320 KB LDS
    const int m0 = blockIdx.y * 16;
    for (int i = threadIdx.x; i < 16 * EMBED; i += 256) {
        int r = i >> 9, c = i & (EMBED - 1);
        Af[i] = features[(size_t)(m0 + r) * EMBED + c];
    }
    __syncthreads();

    const int wave   = threadIdx.x >> 5;
    const int lane   = threadIdx.x & 31;
    const int n_tile = blockIdx.x * 8 + wave;      // wave-uniform guard: EXEC
    if (n_tile * 16 >= NSAMP) return;              // stays all-ones for WMMA

    const int col   = lane & 15;                   // A: M index, B: N index
    const int khalf = (lane >> 4) * 2;             // K sub-pair per lane half

    const int   n    = n_tile * 16 + col;
    const float invn = inv_norm[n];
    const float* wrow = weight + (size_t)index[n] * EMBED;   // gathered row
    const float* arow = Af + col * EMBED;

    __builtin_prefetch(wrow, 0, 1);                // global_prefetch_b8
    __builtin_prefetch(wrow + 256, 0, 1);

    v8f c0 = {}, c1 = {};
    for (int k = 0; k < EMBED; k += 8) {
        v2f a0 = *(const v2f*)(arow + k + khalf);            // ds_load_b64
        v2f b0 = *(const v2f*)(wrow + k + khalf);            // global_load_b64
        c0 = __builtin_amdgcn_wmma_f32_16x16x4_f32(false, a0, false, b0,
                                                   (short)0, c0, false, false);
        v2f a1 = *(const v2f*)(arow + k + 4 + khalf);
        v2f b1 = *(const v2f*)(wrow + k + 4 + khalf);
        c1 = __builtin_amdgcn_wmma_f32_16x16x4_f32(false, a1, false, b1,
                                                   (short)0, c1, false, false);
    }

    // D layout: lane L holds N=L%16, rows M = r + 8*(L>=16) for r=0..7.
    // Same column for all 8 VGPRs -> apply column scale (invn) here, once.
    const int mhi = (lane >> 4) * 8;
    const size_t ncol = (size_t)(n_tile * 16 + col);
    #pragma unroll
    for (int r = 0; r < 8; ++r)
        logits[(size_t)(m0 + mhi + r) * NSAMP + ncol] = (c0[r] + c1[r]) * invn;
}

// ---------------- Kernel 3: loss -------------------------------------------
__global__ void k_zero(float* loss) { if (threadIdx.x == 0) *loss = 0.f; }

__global__ __launch_bounds__(256) void k_loss(const float* __restrict__ logits,
                                              const int* __restrict__ index,
                                              const int* __restrict__ labels,
                                              float* __restrict__ loss) {
    const int b = blockIdx.x;
    const float* row = logits + (size_t)b * NSAMP;
    // online (single-pass) max + sum-of-exp
    float m = -INFINITY, s = 0.f;
    for (int i = threadIdx.x; i < NSAMP; i += 256) {
        float x = row[i];
        if (x > m) { s = s * expf(m - x) + 1.f; m = x; }
        else       { s += expf(x - m); }
    }
    __shared__ float sm[256], ss[256];
    sm[threadIdx.x] = m; ss[threadIdx.x] = s;
    __syncthreads();
    for (int off = 128; off > 0; off >>= 1) {
        if (threadIdx.x < off) {
            float m1 = sm[threadIdx.x], s1 = ss[threadIdx.x];
            float m2 = sm[threadIdx.x + off], s2 = ss[threadIdx.x + off];
            float M = fmaxf(m1, m2);
            sm[threadIdx.x] = M;
            ss[threadIdx.x] = s1 * expf(m1 - M) + s2 * expf(m2 - M);
        }
        __syncthreads();
    }
    if (threadIdx.x == 0) {
        // searchsorted(index, labels[b]) — label guaranteed present
        int lab = labels[b], lo = 0, hi = NSAMP;
        while (lo < hi) { int mid = (lo + hi) >> 1;
                          if (index[mid] < lab) lo = mid + 1; else hi = mid; }
        float logp = row[lo] - sm[0] - logf(ss[0]);
        float val  = fminf(-logp, 69.07755278982137f);   // -log(1e-30) clamp
        atomicAdd(loss, val * (1.0f / (float)BATCH));
    }
}

// ---------------- Launch ----------------------------------------------------
extern "C" void kernel_launch(void* const* d_in, const int* in_sizes, int n_in,
                              void* d_out, int out_size, void* d_ws, size_t ws_size,
                              hipStream_t stream) {
    const float* features = (const float*)d_in[0];
    const float* weight   = (const float*)d_in[1];
    const int*   labels   = (const int*)d_in[2];
    const int*   index    = (const int*)d_in[3];

    float* logits   = (float*)d_out;
    float* loss     = logits + (size_t)BATCH * NSAMP;
    float* inv_norm = (float*)d_ws;                 // 50000 f32 = 200 KB

    k_norms<<<dim3((NSAMP + 7) / 8), 256, 0, stream>>>(weight, index, inv_norm);

    // N tiles: 50000/16 = 3125 (exact); 8 waves/block -> 391 blocks in x
    k_gemm<<<dim3((NSAMP / 16 + 7) / 8, BATCH / 16), 256, 0, stream>>>(
        features, weight, index, inv_norm, logits);

    k_zero<<<1, 64, 0, stream>>>(loss);
    k_loss<<<BATCH, 256, 0, stream>>>(logits, index, labels, loss);
}